// Wavelet1D_78632261255410
// MI455X (gfx1250) — compile-verified
//
#include <hip/hip_runtime.h>
#include <math.h>

typedef __attribute__((ext_vector_type(2))) float v2f;
typedef __attribute__((ext_vector_type(8))) float v8f;

#define TT 4096
#define NN 8192
#define BB 64

// D = A(16x4) * B(4x16) + C, all f32, wave32. 8-arg form per CDNA5 bridge doc.
#define WMMA_F32(a, b, c) \
  __builtin_amdgcn_wmma_f32_16x16x4_f32(false, (a), false, (b), (short)0, (c), false, false)

// A-matrix fragment (16x4 f32): lanes 0-15 -> M rows, VGPR0/1 = K pair, hi half-wave = K+2.
__device__ __forceinline__ v2f afrag(const float* M, int ld, int row0, int k0, int lane) {
  const int m  = row0 + (lane & 15);
  const int kk = k0 + ((lane >> 4) << 1);
  v2f a;
  a.x = M[m * ld + kk];
  a.y = M[m * ld + kk + 1];
  return a;
}
// B-matrix fragment (4x16 f32): lanes 0-15 -> N cols, VGPR0/1 = K pair, hi half-wave = K+2.
__device__ __forceinline__ v2f bfrag(const float* M, int ld, int k0, int col0, int lane) {
  const int n  = col0 + (lane & 15);
  const int kk = k0 + ((lane >> 4) << 1);
  v2f b;
  b.x = M[kk * ld + n];
  b.y = M[(kk + 1) * ld + n];
  return b;
}

// ---------------- Kernel 1: forward FFT of mirror-padded input -----------------
// One workgroup per batch. 8192-pt radix-2 DIT in LDS (64 KB of the 320 KB WGP LDS).
// Emits bins k=1..4096 (the only ones the filter bank keeps) as F[b][j]=FFT[j+1].
__global__ void __launch_bounds__(256) fft_fwd_kernel(const float* __restrict__ x,
                                                      float* __restrict__ Fre,
                                                      float* __restrict__ Fim) {
  extern __shared__ float sm[];
  float* re = sm;
  float* im = sm + NN;
  const int b   = blockIdx.x;
  const int tid = threadIdx.x;
  const float* xb = x + b * TT;

  // Bit-reversed load of the mirror-padded signal: [rev(first 2048) | x | rev(last 2048)]
  for (int i = tid; i < NN; i += 256) {
    const unsigned r = __brev((unsigned)i) >> 19;  // 13-bit reverse
    float v;
    if (r < 2048u)      v = xb[2047 - r];
    else if (r < 6144u) v = xb[r - 2048];
    else                v = xb[4095 - (r - 6144)];
    re[i] = v;
    im[i] = 0.f;
  }
  __syncthreads();

  for (int st = 1; st <= 13; ++st) {
    const int half = 1 << (st - 1);
    for (int id = tid; id < NN / 2; id += 256) {
      const int pos = id & (half - 1);
      const int grp = id >> (st - 1);
      const int i0  = (grp << st) + pos;
      const int i1  = i0 + half;
      float sn, cs;  // forward FFT: e^{-2*pi*i*pos/len}
      __sincosf(-6.283185307179586f * (float)pos / (float)(1 << st), &sn, &cs);
      const float xr = re[i1], xi = im[i1];
      const float tr = xr * cs - xi * sn;
      const float ti = xr * sn + xi * cs;
      re[i1] = re[i0] - tr;
      im[i1] = im[i0] - ti;
      re[i0] += tr;
      im[i0] += ti;
    }
    __syncthreads();
  }

  for (int j = tid; j < TT; j += 256) {
    Fre[b * TT + j] = re[j + 1];
    Fim[b * TT + j] = im[j + 1];
  }
}

// ---------------- Kernel 2: WMMA four-step partial IDFT + log|.| ---------------
// y[t] = (1/n) e^{2pi i t/n} * sum_{j=0}^{4095} H[j] e^{2pi i j t / 8192},
// H[j] = F[b][j+1]*wft[s][j+1]. Split j = j1 + 64*j2, t = t2 + 128*t1:
//   Stage A: G[j1][t2]  = sum_j2 A[j1][j2] e^{2pi i j2 t2/128}      (64x128, K=64)
//   Twiddle: G'         = G .* e^{2pi i j1 t2 / 8192}
//   Stage C: Y[t1][t2]  = sum_j1 e^{2pi i j1 t1/64} G'[j1][t2]      (32x128, K=64; t1 in [16,48))
// The outer e^{2pi i t/n} is unit-modulus -> dropped under abs().
// Output: 0.5*log(|Y|^2) - ln(8192).
__global__ void __launch_bounds__(256) cwt_kernel(const float* __restrict__ Fre,
                                                  const float* __restrict__ Fim,
                                                  const float* __restrict__ wft,
                                                  float* __restrict__ out,
                                                  int S) {
  extern __shared__ float sm[];
  float* Ar = sm;               // 64x64
  float* Ai = Ar + 64 * 64;     // 64x64
  float* Gr = Ai + 64 * 64;     // 64x128
  float* Gi = Gr + 64 * 128;    // 64x128   -> 96 KB total
  const int s    = blockIdx.x;
  const int b    = blockIdx.y;
  const int tid  = threadIdx.x;
  const int lane = tid & 31;    // wave32
  const int wave = tid >> 5;    // 8 waves

  // Build A[j1][j2] = F[b][j+1]*wft[s][j+1], j = j1 + 64*j2 (filter is real).
  const float* fre = Fre + b * TT;
  const float* fim = Fim + b * TT;
  const float* w   = wft + s * NN;
  for (int j = tid; j < TT; j += 256) {
    const float wv = w[j + 1];
    Ar[(j & 63) * 64 + (j >> 6)] = fre[j] * wv;
    Ai[(j & 63) * 64 + (j >> 6)] = fim[j] * wv;
  }
  __syncthreads();

  // ---- Stage A: 4x8 grid of 16x16 tiles, 4 tiles per wave, K=64 in steps of 4.
  // DFT-matrix B fragments generated on the fly (args reduced mod 128 -> [0,2pi)).
  for (int tIdx = wave; tIdx < 32; tIdx += 8) {
    const int row0 = (tIdx >> 3) << 4;  // j1 tile base
    const int col0 = (tIdx & 7) << 4;   // t2 tile base
    v8f gr = {0.f, 0.f, 0.f, 0.f, 0.f, 0.f, 0.f, 0.f};
    v8f gi = {0.f, 0.f, 0.f, 0.f, 0.f, 0.f, 0.f, 0.f};
    const int n   = col0 + (lane & 15);
    const int khi = (lane >> 4) << 1;
    for (int k0 = 0; k0 < 64; k0 += 4) {
      const v2f ar = afrag(Ar, 64, row0, k0, lane);
      const v2f ai = afrag(Ai, 64, row0, k0, lane);
      const int kk = k0 + khi;
      float s0, c0, s1, c1;
      __sincosf((float)((kk * n) & 127) * 0.049087385212340526f, &s0, &c0);
      __sincosf((float)(((kk + 1) * n) & 127) * 0.049087385212340526f, &s1, &c1);
      v2f wr; wr.x = c0;    wr.y = c1;
      v2f wi; wi.x = s0;    wi.y = s1;
      v2f na; na.x = -ai.x; na.y = -ai.y;
      gr = WMMA_F32(ar, wr, gr);   // Gr += Ar*W1r
      gr = WMMA_F32(na, wi, gr);   // Gr -= Ai*W1i
      gi = WMMA_F32(ar, wi, gi);   // Gi += Ar*W1i
      gi = WMMA_F32(ai, wr, gi);   // Gi += Ai*W1r
    }
    const int moff = (lane >> 4) << 3;  // C/D layout: hi half-wave = M+8
    for (int v = 0; v < 8; ++v) {
      const int m = row0 + v + moff;
      Gr[m * 128 + n] = gr[v];
      Gi[m * 128 + n] = gi[v];
    }
  }
  __syncthreads();

  // ---- Twiddle: G *= e^{2pi i j1 t2 / 8192}, in place.
  for (int idx = tid; idx < 64 * 128; idx += 256) {
    const int j1 = idx >> 7;
    const int t2 = idx & 127;
    float sn, cs;
    __sincosf((float)((j1 * t2) & 8191) * 7.669903939428206e-4f, &sn, &cs);
    const float grv = Gr[idx], giv = Gi[idx];
    Gr[idx] = grv * cs - giv * sn;
    Gi[idx] = grv * sn + giv * cs;
  }
  __syncthreads();

  // ---- Stage C: 2x8 grid of tiles (t1' = t1-16 in [0,32)), 2 tiles per wave.
  // Accumulators stay in VGPRs; log-abs epilogue fused, no extra LDS round-trip.
  const float ln_n = 9.010913347279288f;  // ln(8192)
  float* ob = out + ((size_t)b * (size_t)S + (size_t)s) * (size_t)TT;
  for (int tIdx = wave; tIdx < 16; tIdx += 8) {
    const int row0 = (tIdx >> 3) << 4;  // t1' tile base (0 or 16)
    const int col0 = (tIdx & 7) << 4;   // t2 tile base
    v8f yr = {0.f, 0.f, 0.f, 0.f, 0.f, 0.f, 0.f, 0.f};
    v8f yi = {0.f, 0.f, 0.f, 0.f, 0.f, 0.f, 0.f, 0.f};
    const int mrow = row0 + (lane & 15);     // t1' for this lane's A rows
    const int khi  = (lane >> 4) << 1;
    for (int k0 = 0; k0 < 64; k0 += 4) {
      const int kk = k0 + khi;                // j1
      float s0, c0, s1, c1;
      __sincosf((float)((kk * (mrow + 16)) & 63) * 0.09817477042468103f, &s0, &c0);
      __sincosf((float)(((kk + 1) * (mrow + 16)) & 63) * 0.09817477042468103f, &s1, &c1);
      v2f wr; wr.x = c0;  wr.y = c1;
      v2f wi; wi.x = s0;  wi.y = s1;
      v2f nw; nw.x = -s0; nw.y = -s1;
      const v2f gr = bfrag(Gr, 128, k0, col0, lane);
      const v2f gi = bfrag(Gi, 128, k0, col0, lane);
      yr = WMMA_F32(wr, gr, yr);   // Yr += W2r*Gr
      yr = WMMA_F32(nw, gi, yr);   // Yr -= W2i*Gi
      yi = WMMA_F32(wr, gi, yi);   // Yi += W2r*Gi
      yi = WMMA_F32(wi, gr, yi);   // Yi += W2i*Gr
    }
    const int n    = col0 + (lane & 15);
    const int moff = (lane >> 4) << 3;
    for (int v = 0; v < 8; ++v) {
      const int t1p = row0 + v + moff;        // t = t2 + 128*(t1p+16); out idx = t-2048
      const float rr = yr[v], ii = yi[v];
      ob[t1p * 128 + n] = 0.5f * __logf(rr * rr + ii * ii) - ln_n;
    }
  }
}

extern "C" void kernel_launch(void* const* d_in, const int* in_sizes, int n_in,
                              void* d_out, int out_size, void* d_ws, size_t ws_size,
                              hipStream_t stream) {
  const float* inputs = (const float*)d_in[0];   // [64, 4096] f32
  const float* wft    = (const float*)d_in[1];   // [S, 8192]  f32
  float* out = (float*)d_out;                    // [64, S, 4096] f32
  const int S = in_sizes[1] / NN;                // 75 scales

  float* Fre = (float*)d_ws;                     // [64, 4096]
  float* Fim = Fre + (size_t)BB * TT;            // [64, 4096]  (2 MB total scratch)

  fft_fwd_kernel<<<BB, 256, NN * 2 * sizeof(float), stream>>>(inputs, Fre, Fim);
  cwt_kernel<<<dim3(S, BB), 256, (64 * 64 * 2 + 64 * 128 * 2) * sizeof(float), stream>>>(
      Fre, Fim, wft, out, S);
}